// unit_aagcn_68848325755549
// MI455X (gfx1250) — compile-verified
//
#include <hip/hip_runtime.h>
#include <hip/hip_bf16.h>
#include <math.h>

#define N_ 64
#define C_ 64
#define T_ 300
#define V_ 25
#define S_ 3
#define INTER_ 16

typedef float v2f __attribute__((ext_vector_type(2)));
typedef float v8f __attribute__((ext_vector_type(8)));

__device__ __forceinline__ v8f wmma4(v2f a, v2f b, v8f c) {
    // D(16x16,f32) = A(16x4,f32) * B(4x16,f32) + C
    return __builtin_amdgcn_wmma_f32_16x16x4_f32(
        /*neg_a=*/false, a, /*neg_b=*/false, b,
        /*c_mod=*/(short)0, c, /*reuse_a=*/false, /*reuse_b=*/false);
}

__device__ __forceinline__ float sigm(float x) { return 1.f / (1.f + expf(-x)); }

// ---------------------------------------------------------------------------
// K1: per (n,s) compute G[n,s,v,w] = A[s,v,w] + alpha*tanh(M/(INTER*T))
//     M[v,w] = sum_{i,t} (Wa_s x_t + ba)[i,v] * (Wb_s x_t + bb)[i,w]
// ---------------------------------------------------------------------------
__global__ __launch_bounds__(32) void k1_adj(
    const float* __restrict__ x, const float* __restrict__ Aadj,
    const float* __restrict__ alpha,
    const float* __restrict__ Wa, const float* __restrict__ ba,
    const float* __restrict__ Wb, const float* __restrict__ bb,
    float* __restrict__ G)
{
    const int n = blockIdx.x, s = blockIdx.y;
    const int lane = threadIdx.x;
    const int m = lane & 15, h = lane >> 4;

    __shared__ float lWa[16 * 64];
    __shared__ float lWb[16 * 64];
    __shared__ float lx[64 * 32];   // x_t, cols padded to 32 (zeros)
    __shared__ float la[16 * 32];   // a_t (i x v)
    __shared__ float lb[16 * 32];   // b_t (i x v)

    for (int i = lane; i < 16 * 64; i += 32) {
        lWa[i] = Wa[s * 16 * 64 + i];
        lWb[i] = Wb[s * 16 * 64 + i];
    }
    for (int i = lane; i < 64 * 32; i += 32) lx[i] = 0.f;
    __syncthreads();

    v8f accM[2][2];
    for (int a0 = 0; a0 < 2; ++a0)
        for (int b0 = 0; b0 < 2; ++b0) accM[a0][b0] = {};

    for (int t = 0; t < T_; ++t) {
        // stage x[n,:,t,:] into LDS (64 x 25, stride 32)
        for (int i = lane; i < 64 * V_; i += 32) {
            int c = i / V_, v = i % V_;
            lx[c * 32 + v] = x[((size_t)(n * C_ + c) * T_ + t) * V_ + v];
        }
        __syncthreads();

        // a = Wa_s @ x_t, b = Wb_s @ x_t  (16x32 each), K = 64
        v8f accA[2], accB[2];
        accA[0] = {}; accA[1] = {}; accB[0] = {}; accB[1] = {};
        for (int k = 0; k < 64; k += 4) {
            const int kk = k + 2 * h;
            v2f av, bv;
            av.x = lWa[m * 64 + kk]; av.y = lWa[m * 64 + kk + 1];
            bv.x = lWb[m * 64 + kk]; bv.y = lWb[m * 64 + kk + 1];
            for (int nt = 0; nt < 2; ++nt) {
                v2f xv;
                xv.x = lx[kk * 32 + nt * 16 + m];
                xv.y = lx[(kk + 1) * 32 + nt * 16 + m];
                accA[nt] = wmma4(av, xv, accA[nt]);
                accB[nt] = wmma4(bv, xv, accB[nt]);
            }
        }
        __syncthreads();
        // spill a,b to LDS (+bias per row i)
        for (int nt = 0; nt < 2; ++nt)
            for (int r = 0; r < 8; ++r) {
                int row = r + 8 * h;         // i index
                int col = nt * 16 + m;       // v index
                la[row * 32 + col] = accA[nt][r] + ba[s * 16 + row];
                lb[row * 32 + col] = accB[nt][r] + bb[s * 16 + row];
            }
        __syncthreads();
        // M += a^T b   (32x32 tiles, K = INTER = 16)
        for (int k = 0; k < 16; k += 4) {
            const int kk = k + 2 * h;
            for (int mt = 0; mt < 2; ++mt) {
                v2f av;
                av.x = la[kk * 32 + mt * 16 + m];
                av.y = la[(kk + 1) * 32 + mt * 16 + m];
                for (int nt = 0; nt < 2; ++nt) {
                    v2f bv;
                    bv.x = lb[kk * 32 + nt * 16 + m];
                    bv.y = lb[(kk + 1) * 32 + nt * 16 + m];
                    accM[mt][nt] = wmma4(av, bv, accM[mt][nt]);
                }
            }
        }
        __syncthreads();
    }

    const float scale = 1.f / (float)(INTER_ * T_);
    const float al = alpha[0];
    for (int mt = 0; mt < 2; ++mt)
        for (int nt = 0; nt < 2; ++nt)
            for (int r = 0; r < 8; ++r) {
                int v = mt * 16 + r + 8 * h;
                int w = nt * 16 + m;
                if (v < V_ && w < V_) {
                    float mv = tanhf(accM[mt][nt][r] * scale);
                    G[((size_t)(n * S_ + s) * V_ + v) * V_ + w] =
                        Aadj[(s * V_ + v) * V_ + w] + al * mv;
                }
            }
}

// ---------------------------------------------------------------------------
// K2: per (n,t): y1 = relu(bn(sum_s Wd_s @ (x_t G_s) + bd_sum) + x_t) -> d_out
// ---------------------------------------------------------------------------
__global__ __launch_bounds__(32) void k2_main(
    const float* __restrict__ x, const float* __restrict__ G,
    const float* __restrict__ Wd, const float* __restrict__ bd,
    const float* __restrict__ gamma, const float* __restrict__ beta,
    const float* __restrict__ mean, const float* __restrict__ var,
    float* __restrict__ out)
{
    const int t = blockIdx.x, n = blockIdx.y;
    const int lane = threadIdx.x;
    const int m = lane & 15, h = lane >> 4;

    __shared__ float lx[64 * 32];   // x_t (c x v), padded cols
    __shared__ float lG[32 * 32];   // G_s (w x v), zero-padded
    __shared__ float lz[64 * 32];   // z_s (c x v)

    for (int i = lane; i < 64 * 32; i += 32) lx[i] = 0.f;
    for (int i = lane; i < 32 * 32; i += 32) lG[i] = 0.f;
    __syncthreads();
    for (int i = lane; i < 64 * V_; i += 32) {
        int c = i / V_, v = i % V_;
        lx[c * 32 + v] = x[((size_t)(n * C_ + c) * T_ + t) * V_ + v];
    }

    v8f accY[4][2];
    for (int a0 = 0; a0 < 4; ++a0)
        for (int b0 = 0; b0 < 2; ++b0) accY[a0][b0] = {};

    for (int s = 0; s < S_; ++s) {
        __syncthreads();
        for (int i = lane; i < V_ * V_; i += 32) {
            int w = i / V_, v = i % V_;
            lG[w * 32 + v] = G[((size_t)(n * S_ + s) * V_ + w) * V_ + v];
        }
        __syncthreads();
        // GEMM1: z = x_t (64x32) @ G_s (32x32), K = 32 (zero padded)
        for (int mt = 0; mt < 4; ++mt) {
            v8f accZ[2]; accZ[0] = {}; accZ[1] = {};
            for (int k = 0; k < 32; k += 4) {
                const int kk = k + 2 * h;
                v2f av;
                av.x = lx[(mt * 16 + m) * 32 + kk];
                av.y = lx[(mt * 16 + m) * 32 + kk + 1];
                for (int nt = 0; nt < 2; ++nt) {
                    v2f bv;
                    bv.x = lG[kk * 32 + nt * 16 + m];
                    bv.y = lG[(kk + 1) * 32 + nt * 16 + m];
                    accZ[nt] = wmma4(av, bv, accZ[nt]);
                }
            }
            for (int nt = 0; nt < 2; ++nt)
                for (int r = 0; r < 8; ++r)
                    lz[(mt * 16 + r + 8 * h) * 32 + nt * 16 + m] = accZ[nt][r];
        }
        __syncthreads();
        // GEMM2: Y += Wd_s (64x64) @ z (64x32), K = 64
        const float* __restrict__ Wds = Wd + (size_t)s * 64 * 64;
        for (int k = 0; k < 64; k += 4) {
            const int kk = k + 2 * h;
            for (int mt = 0; mt < 4; ++mt) {
                v2f av;
                av.x = Wds[(mt * 16 + m) * 64 + kk];
                av.y = Wds[(mt * 16 + m) * 64 + kk + 1];
                for (int nt = 0; nt < 2; ++nt) {
                    v2f bv;
                    bv.x = lz[kk * 32 + nt * 16 + m];
                    bv.y = lz[(kk + 1) * 32 + nt * 16 + m];
                    accY[mt][nt] = wmma4(av, bv, accY[mt][nt]);
                }
            }
        }
    }

    // epilogue: bd_sum + BN(eval) + residual + relu
    for (int mt = 0; mt < 4; ++mt)
        for (int r = 0; r < 8; ++r) {
            int o = mt * 16 + r + 8 * h;
            float bsum = bd[o] + bd[C_ + o] + bd[2 * C_ + o];
            float invs = gamma[o] * rsqrtf(var[o] + 1e-5f);
            for (int nt = 0; nt < 2; ++nt) {
                int v = nt * 16 + m;
                if (v < V_) {
                    float val = accY[mt][nt][r] + bsum;
                    val = (val - mean[o]) * invs + beta[o];
                    size_t idx = ((size_t)(n * C_ + o) * T_ + t) * V_ + v;
                    val += x[idx];
                    out[idx] = val > 0.f ? val : 0.f;
                }
            }
        }
}

// ---------------------------------------------------------------------------
// K3: ST[n,c,v] = sum_t y1
// ---------------------------------------------------------------------------
__global__ void k3_sumT(const float* __restrict__ y1, float* __restrict__ ST) {
    int i = blockIdx.x * blockDim.x + threadIdx.x;
    if (i >= N_ * C_ * V_) return;
    int v = i % V_, c = (i / V_) % C_, n = i / (V_ * C_);
    const float* p = y1 + ((size_t)(n * C_ + c) * T_) * V_ + v;
    float s = 0.f;
    for (int t = 0; t < T_; ++t) s += p[(size_t)t * V_];
    ST[i] = s;
}

// K4: sa[n,v] = sigmoid(conv_V(ST/T))
__global__ void k4_sa(const float* __restrict__ ST, const float* __restrict__ Wsa,
                      const float* __restrict__ bsa, float* __restrict__ sa) {
    int i = blockIdx.x * blockDim.x + threadIdx.x;
    if (i >= N_ * V_) return;
    int v = i % V_, n = i / V_;
    float acc = bsa[0];
    const float invT = 1.f / (float)T_;
    for (int c = 0; c < C_; ++c) {
        const float* stp = ST + (size_t)(n * C_ + c) * V_;
        for (int k = 0; k < V_; ++k) {
            int vv = v + k - (V_ - 1) / 2;
            if (vv >= 0 && vv < V_) acc += Wsa[c * V_ + k] * stp[vv] * invT;
        }
    }
    sa[i] = sigm(acc);
}

// K5a: SV[n,c,t] = mean_v y1*(1+sa)
__global__ void k5_sv(const float* __restrict__ y1, const float* __restrict__ sa,
                      float* __restrict__ SV) {
    int i = blockIdx.x * blockDim.x + threadIdx.x;
    if (i >= N_ * C_ * T_) return;
    int t = i % T_, c = (i / T_) % C_, n = i / (T_ * C_);
    const float* p = y1 + ((size_t)(n * C_ + c) * T_ + t) * V_;
    const float* sap = sa + n * V_;
    float s = 0.f;
    for (int v = 0; v < V_; ++v) s += p[v] * (1.f + sap[v]);
    SV[i] = s * (1.f / (float)V_);
}

// K5b: ta[n,t] = sigmoid(conv_9(SV))
__global__ void k5_ta(const float* __restrict__ SV, const float* __restrict__ Wta,
                      const float* __restrict__ bta, float* __restrict__ ta) {
    int i = blockIdx.x * blockDim.x + threadIdx.x;
    if (i >= N_ * T_) return;
    int t = i % T_, n = i / T_;
    float acc = bta[0];
    for (int c = 0; c < C_; ++c) {
        const float* svp = SV + (size_t)(n * C_ + c) * T_;
        for (int k = 0; k < 9; ++k) {
            int tt = t + k - 4;
            if (tt >= 0 && tt < T_) acc += Wta[c * 9 + k] * svp[tt];
        }
    }
    ta[i] = sigm(acc);
}

// K6a: SC[n,c] = mean_{t,v} y1*(1+sa)*(1+ta)
__global__ __launch_bounds__(256) void k6_sc(
    const float* __restrict__ y1, const float* __restrict__ sa,
    const float* __restrict__ ta, float* __restrict__ SC) {
    int c = blockIdx.x, n = blockIdx.y;
    __shared__ float red[256];
    const float* p = y1 + (size_t)(n * C_ + c) * T_ * V_;
    float s = 0.f;
    for (int i = threadIdx.x; i < T_ * V_; i += 256) {
        int t = i / V_, v = i % V_;
        s += p[i] * (1.f + sa[n * V_ + v]) * (1.f + ta[n * T_ + t]);
    }
    red[threadIdx.x] = s;
    __syncthreads();
    for (int o = 128; o > 0; o >>= 1) {
        if (threadIdx.x < o) red[threadIdx.x] += red[threadIdx.x + o];
        __syncthreads();
    }
    if (threadIdx.x == 0) SC[n * C_ + c] = red[0] * (1.f / (float)(T_ * V_));
}

// K6b: ca[n,c] = sigmoid(fc2(relu(fc1(SC))))
__global__ __launch_bounds__(64) void k6_ca(
    const float* __restrict__ SC, const float* __restrict__ Wfc1,
    const float* __restrict__ bfc1, const float* __restrict__ Wfc2,
    const float* __restrict__ bfc2, float* __restrict__ ca) {
    int n = blockIdx.x, j = threadIdx.x;
    __shared__ float hbuf[32];
    if (j < 32) {
        float a = bfc1[j];
        for (int c = 0; c < C_; ++c) a += Wfc1[j * C_ + c] * SC[n * C_ + c];
        hbuf[j] = a > 0.f ? a : 0.f;
    }
    __syncthreads();
    float a = bfc2[j];
    for (int k = 0; k < 32; ++k) a += Wfc2[j * 32 + k] * hbuf[k];
    ca[n * C_ + j] = sigm(a);
}

// K7: in-place gate product on d_out
__global__ void k7_final(float* __restrict__ y, const float* __restrict__ sa,
                         const float* __restrict__ ta, const float* __restrict__ ca) {
    size_t i = (size_t)blockIdx.x * blockDim.x + threadIdx.x;
    if (i >= (size_t)N_ * C_ * T_ * V_) return;
    int v = (int)(i % V_);
    size_t r = i / V_;
    int t = (int)(r % T_); r /= T_;
    int c = (int)(r % C_);
    int n = (int)(r / C_);
    y[i] = y[i] * (1.f + sa[n * V_ + v]) * (1.f + ta[n * T_ + t]) * (1.f + ca[n * C_ + c]);
}

// ---------------------------------------------------------------------------
extern "C" void kernel_launch(void* const* d_in, const int* in_sizes, int n_in,
                              void* d_out, int out_size, void* d_ws, size_t ws_size,
                              hipStream_t stream) {
    const float* x     = (const float*)d_in[0];
    const float* Aadj  = (const float*)d_in[1];
    const float* alpha = (const float*)d_in[2];
    const float* Wa    = (const float*)d_in[3];
    const float* ba    = (const float*)d_in[4];
    const float* Wb    = (const float*)d_in[5];
    const float* bb    = (const float*)d_in[6];
    const float* Wd    = (const float*)d_in[7];
    const float* bd    = (const float*)d_in[8];
    const float* bng   = (const float*)d_in[9];
    const float* bnb   = (const float*)d_in[10];
    const float* bnm   = (const float*)d_in[11];
    const float* bnv   = (const float*)d_in[12];
    const float* Wsa   = (const float*)d_in[13];
    const float* bsa   = (const float*)d_in[14];
    const float* Wta   = (const float*)d_in[15];
    const float* bta   = (const float*)d_in[16];
    const float* Wfc1  = (const float*)d_in[17];
    const float* bfc1  = (const float*)d_in[18];
    const float* Wfc2  = (const float*)d_in[19];
    const float* bfc2  = (const float*)d_in[20];
    float* out = (float*)d_out;

    float* ws = (float*)d_ws;
    float* G  = ws;                       // N*S*V*V          = 120000
    float* ST = G  + (size_t)N_ * S_ * V_ * V_;   // N*C*V    = 102400
    float* sa = ST + (size_t)N_ * C_ * V_;        // N*V      = 1600
    float* SV = sa + (size_t)N_ * V_;             // N*C*T    = 1228800
    float* ta = SV + (size_t)N_ * C_ * T_;        // N*T      = 19200
    float* SC = ta + (size_t)N_ * T_;             // N*C      = 4096
    float* ca = SC + (size_t)N_ * C_;             // N*C      = 4096

    k1_adj<<<dim3(N_, S_), 32, 0, stream>>>(x, Aadj, alpha, Wa, ba, Wb, bb, G);
    k2_main<<<dim3(T_, N_), 32, 0, stream>>>(x, G, Wd, bd, bng, bnb, bnm, bnv, out);
    k3_sumT<<<(N_ * C_ * V_ + 255) / 256, 256, 0, stream>>>(out, ST);
    k4_sa<<<(N_ * V_ + 255) / 256, 256, 0, stream>>>(ST, Wsa, bsa, sa);
    k5_sv<<<(N_ * C_ * T_ + 255) / 256, 256, 0, stream>>>(out, sa, SV);
    k5_ta<<<(N_ * T_ + 255) / 256, 256, 0, stream>>>(SV, Wta, bta, ta);
    k6_sc<<<dim3(C_, N_), 256, 0, stream>>>(out, sa, ta, SC);
    k6_ca<<<N_, 64, 0, stream>>>(SC, Wfc1, bfc1, Wfc2, bfc2, ca);
    {
        size_t total = (size_t)N_ * C_ * T_ * V_;
        k7_final<<<(unsigned)((total + 255) / 256), 256, 0, stream>>>(out, sa, ta, ca);
    }
}